// GCNNet_40175124087119
// MI455X (gfx1250) — compile-verified
//
#include <hip/hip_runtime.h>
#include <math.h>

typedef float v2f __attribute__((ext_vector_type(2)));
typedef float v8f __attribute__((ext_vector_type(8)));

#define N_NODES 100000
#define F_IN    512
#define DIM     32
#define NCLS    40
#define NCLS_PAD 48
#define ROW_TILES (N_NODES / 16)   // 6250 exact

// ---------------- degree / normalization ----------------

__global__ void k_init_deg(float* deg) {
    int i = blockIdx.x * 256 + threadIdx.x;
    if (i < N_NODES) deg[i] = 1.0f;   // self-loop contribution
}

__global__ void k_deg_edges(const long long* __restrict__ dst, float* deg, int E) {
    int e = blockIdx.x * 256 + threadIdx.x;
    if (e < E) atomicAdd(&deg[(int)dst[e]], 1.0f);
}

__global__ void k_dinv(float* deg) {
    int i = blockIdx.x * 256 + threadIdx.x;
    if (i < N_NODES) deg[i] = rsqrtf(deg[i]);   // deg >= 1 always
}

// ---------------- GEMM 1: X[100000x512] @ W1[512x32] via fp32 WMMA ----------------
// One wave computes one 16x16 output tile; 8 waves/block = 4 row-tiles x 2 col-tiles.
// W1 (exactly 64 KB) staged in LDS (WGP has 320 KB).

__global__ __launch_bounds__(256)
void k_xw1_wmma(const float* __restrict__ X, const float* __restrict__ W,
                float* __restrict__ out) {
    __shared__ float w_lds[F_IN * DIM];            // 64 KB
    int tid = threadIdx.x;
    for (int i = tid; i < F_IN * DIM; i += 256) w_lds[i] = W[i];
    __syncthreads();

    int wave = tid >> 5;
    int lane = tid & 31;
    int rowTile = blockIdx.x * 4 + (wave >> 1);    // 16-row tile
    int n0 = (wave & 1) * 16;                      // 16-col tile (DIM=32 -> 2 tiles)
    if (rowTile >= ROW_TILES) return;              // wave-uniform: EXEC stays all-1s

    int m0 = rowTile * 16;
    int lh = lane & 15;                            // row / col within tile
    int kh = lane >> 4;                            // K half-select (0 -> K+{0,1}, 1 -> K+{2,3})

    const float* arow = X + (long long)(m0 + lh) * F_IN + 2 * kh;
    const float* brow = w_lds + (2 * kh) * DIM + n0 + lh;

    v8f c = {};
    for (int k = 0; k < F_IN; k += 4) {
        v2f a, b;
        a.x = arow[k];               // A[m0+lh][k + 2*kh]
        a.y = arow[k + 1];           // A[m0+lh][k + 2*kh + 1]
        b.x = brow[k * DIM];         // B[k + 2*kh    ][n0+lh]
        b.y = brow[k * DIM + DIM];   // B[k + 2*kh + 1][n0+lh]
        c = __builtin_amdgcn_wmma_f32_16x16x4_f32(false, a, false, b,
                                                  (short)0, c, false, false);
    }
    // C/D layout: vgpr v <-> row m0 + v + 8*kh, col n0 + lh
    float* orow = out + (long long)(m0 + 8 * kh) * DIM + n0 + lh;
#pragma unroll
    for (int v = 0; v < 8; ++v) orow[v * DIM] = c[v];
}

// ---------------- layer-1 aggregation ----------------

__global__ void k_self_init32(const float* __restrict__ xw, const float* __restrict__ dinv,
                              float* __restrict__ h) {
    int i = blockIdx.x * 256 + threadIdx.x;        // over N_NODES*32 = 3.2M
    if (i < N_NODES * DIM) {
        float d = dinv[i >> 5];
        h[i] = xw[i] * d * d;                      // self-loop term dinv^2
    }
}

__global__ void k_agg32(const float* __restrict__ xw, const long long* __restrict__ src,
                        const long long* __restrict__ dst, const float* __restrict__ dinv,
                        float* __restrict__ h, int E) {
    long long t = (long long)blockIdx.x * 256 + threadIdx.x;
    long long e = t >> 5;                          // one wave per edge (scalar src/dst/norm)
    int f = (int)(t & 31);
    if (e < (long long)E) {
        int s = (int)src[e], d = (int)dst[e];
        float norm = dinv[s] * dinv[d];
        atomicAdd(&h[(long long)d * DIM + f], xw[(long long)s * DIM + f] * norm);
    }
}

__global__ void k_relu_bias(float* h, const float* __restrict__ b) {
    int i = blockIdx.x * 256 + threadIdx.x;
    if (i < N_NODES * DIM) {
        float v = h[i] + b[i & 31];
        h[i] = v > 0.0f ? v : 0.0f;
    }
}

// ---------------- GEMM 2: H[100000x32] @ W2[32x40] via fp32 WMMA ----------------
// Cols padded 40 -> 48 (3 tiles) in LDS with zeros; guarded stores.

__global__ __launch_bounds__(256)
void k_xw2_wmma(const float* __restrict__ H, const float* __restrict__ W,
                float* __restrict__ out) {
    __shared__ float w_lds[DIM * NCLS_PAD];        // 6 KB
    int tid = threadIdx.x;
    for (int i = tid; i < DIM * NCLS_PAD; i += 256) {
        int kk = i / NCLS_PAD, n = i % NCLS_PAD;
        w_lds[i] = (n < NCLS) ? W[kk * NCLS + n] : 0.0f;
    }
    __syncthreads();

    int wave = tid >> 5;
    int lane = tid & 31;
    int rowTile = blockIdx.x * 8 + wave;
    if (rowTile >= ROW_TILES) return;

    int m0 = rowTile * 16;
    int lh = lane & 15;
    int kh = lane >> 4;
    const float* arow = H + (long long)(m0 + lh) * DIM + 2 * kh;

    for (int nt = 0; nt < 3; ++nt) {
        int n0 = nt * 16;
        const float* brow = w_lds + (2 * kh) * NCLS_PAD + n0 + lh;
        v8f c = {};
#pragma unroll
        for (int k = 0; k < DIM; k += 4) {
            v2f a, b;
            a.x = arow[k];
            a.y = arow[k + 1];
            b.x = brow[k * NCLS_PAD];
            b.y = brow[k * NCLS_PAD + NCLS_PAD];
            c = __builtin_amdgcn_wmma_f32_16x16x4_f32(false, a, false, b,
                                                      (short)0, c, false, false);
        }
        int col = n0 + lh;
        if (col < NCLS) {
            float* orow = out + (long long)(m0 + 8 * kh) * NCLS + col;
#pragma unroll
            for (int v = 0; v < 8; ++v) orow[v * NCLS] = c[v];
        }
    }
}

// ---------------- layer-2 aggregation + log_softmax ----------------

__global__ void k_self_init40(const float* __restrict__ z, const float* __restrict__ dinv,
                              float* __restrict__ out) {
    int i = blockIdx.x * 256 + threadIdx.x;        // over N_NODES*40 = 4M
    if (i < N_NODES * NCLS) {
        float d = dinv[i / NCLS];
        out[i] = z[i] * d * d;
    }
}

__global__ void k_agg40(const float* __restrict__ z, const long long* __restrict__ src,
                        const long long* __restrict__ dst, const float* __restrict__ dinv,
                        float* __restrict__ out, int E) {
    long long t = (long long)blockIdx.x * 256 + threadIdx.x;
    long long e = t >> 6;                          // 64 threads per edge, 40 active
    int f = (int)(t & 63);
    if (e < (long long)E && f < NCLS) {
        int s = (int)src[e], d = (int)dst[e];
        float norm = dinv[s] * dinv[d];
        atomicAdd(&out[(long long)d * NCLS + f], z[(long long)s * NCLS + f] * norm);
    }
}

__global__ void k_log_softmax(float* __restrict__ out, const float* __restrict__ b2) {
    int i = blockIdx.x * 256 + threadIdx.x;
    if (i < N_NODES) {
        float* row = out + (long long)i * NCLS;
        float v[NCLS];
        float m = -INFINITY;
#pragma unroll
        for (int c = 0; c < NCLS; ++c) { v[c] = row[c] + b2[c]; m = fmaxf(m, v[c]); }
        float s = 0.0f;
#pragma unroll
        for (int c = 0; c < NCLS; ++c) s += __expf(v[c] - m);
        float l = __logf(s);
#pragma unroll
        for (int c = 0; c < NCLS; ++c) row[c] = v[c] - m - l;
    }
}

// ---------------- launch ----------------

extern "C" void kernel_launch(void* const* d_in, const int* in_sizes, int n_in,
                              void* d_out, int out_size, void* d_ws, size_t ws_size,
                              hipStream_t stream) {
    const float*     X   = (const float*)d_in[0];
    const long long* EI  = (const long long*)d_in[1];   // int64 edge_index [2, E]
    const float*     W1  = (const float*)d_in[2];
    const float*     B1  = (const float*)d_in[3];
    const float*     W2  = (const float*)d_in[4];
    const float*     B2  = (const float*)d_in[5];
    float*           OUT = (float*)d_out;

    int E = in_sizes[1] / 2;
    const long long* src = EI;
    const long long* dst = EI + E;

    // workspace partition (floats): dinv | xw1 | h | z2  -> 10.5M floats = 42 MB
    float* ws   = (float*)d_ws;
    float* dinv = ws;                       // N_NODES
    float* xw1  = ws + 100000;              // N_NODES*32
    float* h    = xw1 + (size_t)N_NODES * DIM;
    float* z2   = h   + (size_t)N_NODES * DIM;

    int bN    = (N_NODES + 255) / 256;
    int bE    = (E + 255) / 256;
    int bNF1  = (N_NODES * DIM + 255) / 256;
    int bNF2  = (N_NODES * NCLS + 255) / 256;
    int bE32  = (int)(((long long)E * 32 + 255) / 256);
    int bE64  = (int)(((long long)E * 64 + 255) / 256);

    // degrees -> dinv
    k_init_deg <<<bN, 256, 0, stream>>>(dinv);
    k_deg_edges<<<bE, 256, 0, stream>>>(dst, dinv, E);
    k_dinv     <<<bN, 256, 0, stream>>>(dinv);

    // layer 1
    k_xw1_wmma   <<<(ROW_TILES + 3) / 4, 256, 0, stream>>>(X, W1, xw1);
    k_self_init32<<<bNF1, 256, 0, stream>>>(xw1, dinv, h);
    k_agg32      <<<bE32, 256, 0, stream>>>(xw1, src, dst, dinv, h, E);
    k_relu_bias  <<<bNF1, 256, 0, stream>>>(h, B1);

    // layer 2
    k_xw2_wmma   <<<(ROW_TILES + 7) / 8, 256, 0, stream>>>(h, W2, z2);
    k_self_init40<<<bNF2, 256, 0, stream>>>(z2, dinv, OUT);
    k_agg40      <<<bE64, 256, 0, stream>>>(z2, src, dst, dinv, OUT, E);
    k_log_softmax<<<bN, 256, 0, stream>>>(OUT, B2);
}